// ClusterTverskyLoss_2800318677065
// MI455X (gfx1250) — compile-verified
//
#include <hip/hip_runtime.h>
#include <hip/hip_bf16.h>

// ---------------- problem constants (from reference) ----------------
#define HH    2048
#define WW    2048
#define BSZ   32
#define GG    64               // HH/BSZ
#define NSEG  4097             // GG*GG + 1
#define EPSF  1e-6f

// ---------------- kernel geometry ----------------
#define WAVES_PER_WG   2
#define WG_THREADS     (WAVES_PER_WG * 32)
#define TILES_PER_WAVE 4
#define TILE_BYTES     4096                    // 32*32*4B
#define BUF_BYTES      (3 * TILE_BYTES)        // pred + target + seg
#define WAVE_LDS       (2 * BUF_BYTES)         // double buffered = 24 KB
#define SMEM_BYTES     (WAVES_PER_WG * WAVE_LDS)

// 16-byte int vector + address-space-qualified pointers (per hipcc diagnostic,
// the async builtin takes v4i pointers in AS1/AS3)
typedef int vi4 __attribute__((vector_size(16)));
typedef __attribute__((address_space(1))) vi4* g_vi4p;
typedef __attribute__((address_space(3))) vi4* l_vi4p;

// ---------------- wave32 helpers ----------------
__device__ __forceinline__ unsigned laneId() {
  return __builtin_amdgcn_mbcnt_lo(~0u, 0u);   // wave32: lane id
}

__device__ __forceinline__ unsigned ballot32(bool p) {
#if __has_builtin(__builtin_amdgcn_ballot_w32)
  return __builtin_amdgcn_ballot_w32(p);
#else
  return (unsigned)__ballot((int)p);
#endif
}

// 32-lane xor-butterfly sum via ds_bpermute; result in every lane
__device__ __forceinline__ float waveSum(float v, unsigned lane) {
#pragma unroll
  for (int m = 16; m; m >>= 1) {
    int idx = (int)((lane ^ (unsigned)m) << 2);
    v += __int_as_float(__builtin_amdgcn_ds_bpermute(idx, __float_as_int(v)));
  }
  return v;
}

// ---------------- CDNA5 async global -> LDS (16B per lane) ----------------
__device__ __forceinline__ void asyncLoad16(const void* g, void* l) {
#if __has_builtin(__builtin_amdgcn_global_load_async_to_lds_b128)
  __builtin_amdgcn_global_load_async_to_lds_b128((g_vi4p)g, (l_vi4p)l, 0, 0);
#else
  unsigned loff = (unsigned)(__SIZE_TYPE__)(__attribute__((address_space(3))) void*)l;
  unsigned long long gaddr = (unsigned long long)(__SIZE_TYPE__)g;
  asm volatile("global_load_async_to_lds_b128 %0, %1, off"
               :: "v"(loff), "v"(gaddr) : "memory");
#endif
}

__device__ __forceinline__ void waitAsyncLe24() {
#if __has_builtin(__builtin_amdgcn_s_wait_asynccnt)
  __builtin_amdgcn_s_wait_asynccnt(24);
#else
  asm volatile("s_wait_asynccnt 24" ::: "memory");
#endif
  asm volatile("" ::: "memory");
}

__device__ __forceinline__ void waitAsync0() {
#if __has_builtin(__builtin_amdgcn_s_wait_asynccnt)
  __builtin_amdgcn_s_wait_asynccnt(0);
#else
  asm volatile("s_wait_asynccnt 0" ::: "memory");
#endif
  asm volatile("" ::: "memory");
}

// ---------------- tile staging: 24 async b128 copies per tile ----------------
__device__ __forceinline__ void issueTile(const float* pred, const float* target,
                                          const int* seg, int t, char* wbuf,
                                          unsigned lane) {
  int b  = t >> 12;          // 4096 tiles per image
  int r  = t & 4095;
  int by = r >> 6;
  int bx = r & 63;
  size_t origin = ((size_t)b * HH + (size_t)by * BSZ) * WW + (size_t)bx * BSZ;
  const char* gp = (const char*)(pred   + origin);
  const char* gt = (const char*)(target + origin);
  const char* gs = (const char*)(seg    + origin);
  unsigned vrow  = lane >> 3;              // 8 lanes cover one 128B tile row
  unsigned vcol  = (lane & 7u) * 16u;      // byte offset within row
  unsigned lbase = lane * 16u;
#pragma unroll
  for (int i = 0; i < 8; i++) {
    unsigned goff = (unsigned)(i * 4 + vrow) * (WW * 4) + vcol;  // bytes
    unsigned loff = (unsigned)i * 512u + lbase;
    asyncLoad16(gp + goff, wbuf + 0 * TILE_BYTES + loff);
    asyncLoad16(gt + goff, wbuf + 1 * TILE_BYTES + loff);
    asyncLoad16(gs + goff, wbuf + 2 * TILE_BYTES + loff);
  }
}

// ---------------- group-by-segment flush (wave-uniform) ----------------
__device__ __forceinline__ void flushSlot(float* acc4, int bIdx, int key,
                                          float spt, float sp, float st, float sc,
                                          unsigned lane) {
  unsigned rem = ballot32(key >= 0);
  while (rem) {
    int ld = __builtin_ctz(rem);
    int ks = __builtin_amdgcn_readlane(key, ld);
    bool mine = (key == ks);
    unsigned m = ballot32(mine);
    float vpt = waveSum(mine ? spt : 0.f, lane);
    float vp  = waveSum(mine ? sp  : 0.f, lane);
    float vt  = waveSum(mine ? st  : 0.f, lane);
    float vc  = waveSum(mine ? sc  : 0.f, lane);
    if ((int)lane == ld && (unsigned)ks < (unsigned)NSEG) {
      float* a = acc4 + ((size_t)bIdx * NSEG + (size_t)ks) * 4;
      atomicAdd(a + 0, vpt);
      atomicAdd(a + 1, vp);
      atomicAdd(a + 2, vt);
      atomicAdd(a + 3, vc);
    }
    rem &= ~m;
  }
}

// ---------------- kernels ----------------
extern "C" __global__ void ctl_zero(float* p, int n) {
  int i = blockIdx.x * blockDim.x + threadIdx.x;
  if (i < n) p[i] = 0.f;
}

extern "C" __global__ void __launch_bounds__(WG_THREADS)
ctl_main(const float* __restrict__ pred, const float* __restrict__ target,
         const int* __restrict__ seg, float* __restrict__ acc4, int totalTiles) {
  extern __shared__ char smem[];
  unsigned lane = laneId();
  unsigned wv   = threadIdx.x >> 5;
  char* wls = smem + wv * WAVE_LDS;

  long gw = (long)blockIdx.x * WAVES_PER_WG + wv;
  int  t0 = (int)(gw * TILES_PER_WAVE);

  if (t0 < totalTiles) issueTile(pred, target, seg, t0, wls, lane);

  for (int k = 0; k < TILES_PER_WAVE; k++) {
    int t = t0 + k;
    if (t >= totalTiles) break;
    bool more = (k + 1 < TILES_PER_WAVE) && (t + 1 < totalTiles);
    if (more) issueTile(pred, target, seg, t + 1, wls + ((k + 1) & 1) * BUF_BYTES, lane);
    if (more) waitAsyncLe24(); else waitAsync0();   // oldest 24 (this tile) done

    const char* buf = wls + (k & 1) * BUF_BYTES;
    const float* lp = (const float*)(buf + 0 * TILE_BYTES);
    const float* lt = (const float*)(buf + 1 * TILE_BYTES);
    const int*   ls = (const int*)  (buf + 2 * TILE_BYTES);
    int b = t >> 12;

    // per-lane 2-slot keyed accumulator down this lane's tile column
    int k0 = -1, k1 = -1;
    float a0pt = 0.f, a0p = 0.f, a0t = 0.f, a0c = 0.f;
    float a1pt = 0.f, a1p = 0.f, a1t = 0.f, a1c = 0.f;
#pragma unroll 4
    for (int r = 0; r < 32; r++) {
      float p  = lp[r * 32 + lane];
      float tt = lt[r * 32 + lane];
      int   s  = ls[r * 32 + lane];
      float pt = p * tt;
      if (s == k0)      { a0pt += pt; a0p += p; a0t += tt; a0c += 1.f; }
      else if (s == k1) { a1pt += pt; a1p += p; a1t += tt; a1c += 1.f; }
      else if (k0 < 0)  { k0 = s; a0pt = pt; a0p = p; a0t = tt; a0c = 1.f; }
      else if (k1 < 0)  { k1 = s; a1pt = pt; a1p = p; a1t = tt; a1c = 1.f; }
      else if ((unsigned)s < (unsigned)NSEG) {       // generic fallback (rare)
        float* a = acc4 + ((size_t)b * NSEG + (size_t)s) * 4;
        atomicAdd(a + 0, pt); atomicAdd(a + 1, p);
        atomicAdd(a + 2, tt); atomicAdd(a + 3, 1.f);
      }
    }
    flushSlot(acc4, b, k0, a0pt, a0p, a0t, a0c, lane);
    flushSlot(acc4, b, k1, a1pt, a1p, a1t, a1c, lane);
    // all LDS reads of this buffer are consumed; drain DS before it is reused
    asm volatile("s_wait_dscnt 0" ::: "memory");
  }
}

extern "C" __global__ void __launch_bounds__(256)
ctl_final(const float* __restrict__ acc4, float* __restrict__ out, int B_) {
  __shared__ float r1[256];
  __shared__ float r2[256];
  int tid = threadIdx.x;
  float total = 0.f;
  for (int b = 0; b < B_; b++) {
    float ls = 0.f, ln = 0.f;
    for (int i = 1 + tid; i < NSEG; i += 256) {
      const float* a = acc4 + ((size_t)b * NSEG + (size_t)i) * 4;
      float cnt = a[3];
      if (cnt > 0.f) {
        float inter = a[0], sp = a[1], st = a[2];
        float fp = cnt - st;            // sum(1 - t)
        float fn = cnt - sp;            // sum(1 - p)
        ls += (inter + EPSF) / (inter + fp + fn + EPSF);
        ln += 1.f;
      }
    }
    r1[tid] = ls; r2[tid] = ln;
    __syncthreads();
    for (int s = 128; s > 0; s >>= 1) {
      if (tid < s) { r1[tid] += r1[tid + s]; r2[tid] += r2[tid + s]; }
      __syncthreads();
    }
    if (tid == 0) total += (r2[0] > 0.f) ? (1.f - r1[0] / r2[0]) : 1.f;
    __syncthreads();
  }
  if (tid == 0) out[0] = total / (float)B_;
}

// ---------------- host launcher ----------------
extern "C" void kernel_launch(void* const* d_in, const int* in_sizes, int n_in,
                              void* d_out, int out_size, void* d_ws, size_t ws_size,
                              hipStream_t stream) {
  (void)n_in; (void)out_size; (void)ws_size;
  const float* pred   = (const float*)d_in[0];
  const float* target = (const float*)d_in[1];
  const int*   seg    = (const int*)d_in[2];
  float*       acc4   = (float*)d_ws;
  float*       outp   = (float*)d_out;

  int B_ = in_sizes[0] / (HH * WW);            // 4 for the reference inputs
  int nAcc = B_ * NSEG * 4;

  ctl_zero<<<(nAcc + 255) / 256, 256, 0, stream>>>(acc4, nAcc);

  int totalTiles = B_ * GG * GG;               // 16384
  int waves = (totalTiles + TILES_PER_WAVE - 1) / TILES_PER_WAVE;
  int wgs   = (waves + WAVES_PER_WG - 1) / WAVES_PER_WG;
  ctl_main<<<wgs, WG_THREADS, SMEM_BYTES, stream>>>(pred, target, seg, acc4, totalTiles);

  ctl_final<<<1, 256, 0, stream>>>(acc4, outp, B_);
}